// UniGCNIIConv_30253749633199
// MI455X (gfx1250) — compile-verified
//
#include <hip/hip_runtime.h>
#include <hip/hip_bf16.h>
#include <math.h>

#define DFEAT 96

typedef __attribute__((ext_vector_type(2))) float v2f;
typedef __attribute__((ext_vector_type(8))) float v8f;

// ---------------------------------------------------------------- zero scratch
__global__ __launch_bounds__(256) void zero_kernel(float* __restrict__ p, long n) {
  long i = (long)blockIdx.x * blockDim.x + threadIdx.x;
  long stride = (long)gridDim.x * blockDim.x;
  for (; i < n; i += stride) p[i] = 0.0f;
}

// ------------------------------------------------- V->E hop: scatter-add to edges
// one wave32 per incidence pair; each lane handles 3 of 96 features
__global__ __launch_bounds__(256) void scatter_v2e(const float* __restrict__ X,
                                                   const int* __restrict__ vertex,
                                                   const int* __restrict__ edges,
                                                   float* __restrict__ esum,
                                                   float* __restrict__ cnt, int M) {
  int tid = blockIdx.x * blockDim.x + threadIdx.x;
  int pair = tid >> 5;
  int lane = tid & 31;
  if (pair >= M) return;
  int v = vertex[pair];
  int e = edges[pair];
  const float* src = X + (long)v * DFEAT;
  float* dst = esum + (long)e * DFEAT;
#pragma unroll
  for (int i = 0; i < 3; ++i)
    unsafeAtomicAdd(dst + lane + 32 * i, src[lane + 32 * i]);
  if (lane == 0) unsafeAtomicAdd(cnt + e, 1.0f);
}

// ------------------------------------------- edge finalize: mean + degE scaling
__global__ __launch_bounds__(256) void edge_finalize(float* __restrict__ Xe,
                                                     const float* __restrict__ cnt,
                                                     const float* __restrict__ degE,
                                                     long total) {
  long i = (long)blockIdx.x * blockDim.x + threadIdx.x;
  if (i >= total) return;
  long e = i / DFEAT;
  float scale = degE[e] / fmaxf(cnt[e], 1.0f);
  Xe[i] *= scale;
}

// ------------------------------------------------- E->V hop: scatter-add to nodes
__global__ __launch_bounds__(256) void scatter_e2v(const float* __restrict__ Xe,
                                                   const int* __restrict__ vertex,
                                                   const int* __restrict__ edges,
                                                   float* __restrict__ Xv, int M) {
  int tid = blockIdx.x * blockDim.x + threadIdx.x;
  int pair = tid >> 5;
  int lane = tid & 31;
  if (pair >= M) return;
  int v = vertex[pair];
  int e = edges[pair];
  const float* src = Xe + (long)e * DFEAT;
  float* dst = Xv + (long)v * DFEAT;
#pragma unroll
  for (int i = 0; i < 3; ++i)
    unsafeAtomicAdd(dst + lane + 32 * i, src[lane + 32 * i]);
}

// ----------------- node finalize: degV scale, row L2 normalize, GCNII residual
// one wave32 per node; Xi written in place over Xv
__global__ __launch_bounds__(256) void node_finalize(float* __restrict__ XvXi,
                                                     const float* __restrict__ X0,
                                                     const float* __restrict__ degV,
                                                     const float* __restrict__ alpha_p,
                                                     int N) {
  int tid = blockIdx.x * blockDim.x + threadIdx.x;
  int node = tid >> 5;
  int lane = tid & 31;
  if (node >= N) return;
  float dv = degV[node];
  float alpha = *alpha_p;
  float* row = XvXi + (long)node * DFEAT;
  float vals[3];
  float ss = 0.0f;
#pragma unroll
  for (int i = 0; i < 3; ++i) {
    float x = row[lane + 32 * i] * dv;
    vals[i] = x;
    ss += x * x;
  }
#pragma unroll
  for (int off = 16; off > 0; off >>= 1) ss += __shfl_xor(ss, off, 32);
  float inv = (ss > 0.0f) ? (1.0f / sqrtf(ss)) : 0.0f;
  const float* x0 = X0 + (long)node * DFEAT;
#pragma unroll
  for (int i = 0; i < 3; ++i)
    row[lane + 32 * i] = (1.0f - alpha) * vals[i] * inv + alpha * x0[lane + 32 * i];
}

// --------------------------- out = (1-beta)*Xi + beta * Xi @ W^T  (f32 WMMA)
// one wave32 per 16x16 output tile; 6 waves/block cover one 16-row block.
// A(16x4 f32) layout: lane<16 -> K=k0,k0+1 ; lane>=16 -> K=k0+2,k0+3 (M = lane%16)
// B(4x16 f32) layout: same K split, N = lane%16 ; B[k][n] = W[n][k]
// C/D(16x16 f32): VGPR r -> row r (lanes 0-15) / row r+8 (lanes 16-31), N = lane%16
__global__ __launch_bounds__(192) void gemm_wmma(const float* __restrict__ Xi,
                                                 const float* __restrict__ W,
                                                 const float* __restrict__ beta_p,
                                                 float* __restrict__ out, int N) {
  int lane = threadIdx.x & 31;
  int wave = threadIdx.x >> 5;      // 0..5 -> output column tile
  int rowBase = blockIdx.x * 16;
  int colBase = wave * 16;
  float beta = *beta_p;

  int half = (lane >= 16) ? 2 : 0;
  int l16 = lane & 15;
  int arow_i = rowBase + l16;
  if (arow_i >= N) arow_i = N - 1;  // clamp loads for ragged tail (stores guarded)
  const float* arow = Xi + (long)arow_i * DFEAT;
  const float* brow = W + (long)(colBase + l16) * DFEAT;

  v8f c = {};
#pragma unroll
  for (int k0 = 0; k0 < DFEAT; k0 += 4) {
    v2f a = *(const v2f*)(arow + k0 + half);   // 8B-aligned: k0+half is even
    v2f b = *(const v2f*)(brow + k0 + half);
    c = __builtin_amdgcn_wmma_f32_16x16x4_f32(
        /*neg_a=*/false, a, /*neg_b=*/false, b,
        /*c_mod=*/(short)0, c, /*reuse_a=*/false, /*reuse_b=*/false);
  }

  float ob = 1.0f - beta;
  int mofs = (lane >= 16) ? 8 : 0;
#pragma unroll
  for (int r = 0; r < 8; ++r) {
    int m = rowBase + r + mofs;
    if (m < N) {
      int n = colBase + l16;
      out[(long)m * DFEAT + n] = ob * Xi[(long)m * DFEAT + n] + beta * c[r];
    }
  }
}

extern "C" void kernel_launch(void* const* d_in, const int* in_sizes, int n_in,
                              void* d_out, int out_size, void* d_ws, size_t ws_size,
                              hipStream_t stream) {
  const float* X      = (const float*)d_in[0];
  const float* X0     = (const float*)d_in[1];
  const float* W      = (const float*)d_in[2];
  const float* degE   = (const float*)d_in[3];
  const float* degV   = (const float*)d_in[4];
  const int*   vertex = (const int*)d_in[5];
  const int*   edges  = (const int*)d_in[6];
  const float* alpha  = (const float*)d_in[7];
  const float* beta   = (const float*)d_in[8];
  float* out = (float*)d_out;

  const int E = in_sizes[3];   // degE has E elements
  const int N = in_sizes[4];   // degV has N elements
  const int M = in_sizes[5];   // incidence pairs

  // workspace layout: esum/Xe [E*96] | cnt [E] | Xv/Xi [N*96]   (~29 MB)
  float* esum = (float*)d_ws;
  float* cnt  = esum + (long)E * DFEAT;
  float* Xv   = cnt + E;

  long ztotal = (long)E * DFEAT + E + (long)N * DFEAT;
  int zgrid = (int)((ztotal + 255) / 256);
  if (zgrid > 65535) zgrid = 65535;
  zero_kernel<<<zgrid, 256, 0, stream>>>(esum, ztotal);

  int pgrid = (int)(((long)M * 32 + 255) / 256);
  scatter_v2e<<<pgrid, 256, 0, stream>>>(X, vertex, edges, esum, cnt, M);

  long etotal = (long)E * DFEAT;
  edge_finalize<<<(int)((etotal + 255) / 256), 256, 0, stream>>>(esum, cnt, degE, etotal);

  scatter_e2v<<<pgrid, 256, 0, stream>>>(esum, vertex, edges, Xv, M);

  node_finalize<<<(N + 7) / 8, 256, 0, stream>>>(Xv, X0, degV, alpha, N);

  gemm_wmma<<<(N + 15) / 16, 192, 0, stream>>>(Xv, W, beta, out, N);
}